// DiscreteHawkes_61856118997059
// MI455X (gfx1250) — compile-verified
//
#include <hip/hip_runtime.h>
#include <hip/hip_bf16.h>
#include <math.h>

// ---------------------------------------------------------------------------
// Discrete Hawkes conditional intensity, MI455X (gfx1250, wave32).
//
// P[t,s] = sum_{tp<t} obs[tp,s] e^{-b(t-tp)} is a linear recurrence ->
// chunked scan with C = 64 time steps per chunk:
//   intra-chunk : local = L @ O  (L = 64x64 strictly-lower decay matrix,
//                 O = 64x256 obs chunk)   via V_WMMA_F32_16X16X4_F32
//   inter-chunk : carry[c+1] = carry[c]*e^{-64b} + E[c]   (64-step scan)
//   query       : P = local[t,s] + carry[t/64,s]*e^{-b*(t%64)}
// Total traffic ~13 MB (L2-resident on 192MB L2); GEMM = 134 MFLOP fp32.
// ---------------------------------------------------------------------------

#define N_TIME  4096
#define N_SPACE 256
#define BATCH   65536
#define CHUNK   64
#define NCHUNK  (N_TIME / CHUNK)   // 64
#define OBS_LD  68                 // padded LDS row stride (floats)

typedef __attribute__((ext_vector_type(2))) float v2f;
typedef __attribute__((ext_vector_type(8))) float v8f;

// ---------------------------------------------------------------------------
// Kernel 1: per-chunk triangular GEMM
//   local[chunk*64 + i, s] = sum_{j<i} e^{-b(i-j)} * obs[chunk*64 + j, s]
// Grid: (NCHUNK, N_SPACE/64).  Block: 512 threads = 16 waves; wave w owns
// output tile (mt = w>>2, nt = w&3), accumulating K=64 in steps of 4.
//
// Fragment layouts per CDNA5 ISA 7.12.2 (wave32):
//   A 16x4 : lane&15 = M row; VGPR0 = K{0|2}, VGPR1 = K{1|3}
//   B 4x16 : lane&15 = N col; VGPR0 = K{0|2}, VGPR1 = K{1|3}
//   C/D    : 8 VGPRs; lane&15 = N col; row M = v + 8*(lane>=16)
//
// Branch-free weights, REVERSED table so fragment (.x,.y) reads ascending
// adjacent addresses (no register-pair swap before the WMMA):
//   sWr[j] = e^{-b*(63-j)} for j <= 62, else 0     (w(d) = sWr[63-d])
//   .x = w(rowA-k0)   = sWr[63 - rowA + k0]
//   .y = w(rowA-k0-1) = sWr[64 - rowA + k0]  (next address)
//
// Obs chunk staged once into LDS, transposed + int->f32 converted:
//   sObs[col*68 + row].  B-frag .x/.y are rows k0,k0+1 at one col ->
//   one aligned ds_load_b64.  Stride 68 (=4 banks) makes the low half-wave
//   (rows k0,k0+1) and high half-wave (rows k0+2,k0+3) hit disjoint bank
//   groups (4c+k0 vs 4c+k0+2 mod 64).
// ---------------------------------------------------------------------------
__global__ __launch_bounds__(512) void hawkes_chunk_gemm(
    const int* __restrict__ obs,      // [N_TIME, N_SPACE]
    const float* __restrict__ beta,   // [1]
    float* __restrict__ localP)       // [N_TIME, N_SPACE]
{
    __shared__ float sWr[128];                // reversed decay weight table
    __shared__ float sObs[64 * OBS_LD];       // transposed f32 obs chunk

    const float b = beta[0];
    if (threadIdx.x < 128) {
        const int j = threadIdx.x;
        sWr[j] = (j <= 62) ? expf(-b * (float)(63 - j)) : 0.f;
    }

    const int chunk     = blockIdx.x;         // time chunk
    const int scol0     = blockIdx.y * 64;    // first obs column of block
    const int chunkBase = chunk * CHUNK;      // first time row of chunk

    // Stage obs[chunkBase + r][scol0 + c] -> sObs[c*OBS_LD + r] as f32.
    // Consecutive threads read consecutive global columns (coalesced).
    for (int e = threadIdx.x; e < 64 * 64; e += 512) {
        const int r = e >> 6;
        const int c = e & 63;
        sObs[c * OBS_LD + r] =
            (float)obs[(chunkBase + r) * N_SPACE + scol0 + c];
    }
    __syncthreads();

    const int wave = threadIdx.x >> 5;        // 0..15
    const int lane = threadIdx.x & 31;
    const int mt   = wave >> 2;               // M tile 0..3
    const int nt   = wave & 3;                // N tile 0..3
    const int lh   = lane & 15;
    const int hb   = lane >> 4;

    const int rowA   = mt * 16 + lh;          // i within chunk (0..63)
    const int colLoc = nt * 16 + lh;          // local column (0..63)

    v8f acc = {0.f, 0.f, 0.f, 0.f, 0.f, 0.f, 0.f, 0.f};

    #pragma unroll
    for (int kt = 0; kt < 16; ++kt) {
        const int k0 = kt * 4 + hb * 2;       // this lane's K for .x

        // A fragment: ascending adjacent pair from reversed weight table.
        const int j = 63 - rowA + k0;         // in [0, 125]
        v2f a;
        a.x = sWr[j];                         // w(rowA - k0)
        a.y = sWr[j + 1];                     // w(rowA - k0 - 1)

        // B fragment: rows k0,k0+1 at this col (aligned 8B LDS load).
        const v2f bv =
            *reinterpret_cast<const v2f*>(&sObs[colLoc * OBS_LD + k0]);

        // (neg_a, A, neg_b, B, c_mod, C, reuse_a, reuse_b)
        acc = __builtin_amdgcn_wmma_f32_16x16x4_f32(
            false, a, false, bv, (short)0, acc, false, false);
    }

    // Store D per C/D layout: VGPR v -> row v + 8*hb of this 16x16 tile.
    const int rowBase = chunkBase + mt * 16 + hb * 8;
    const int outCol  = scol0 + colLoc;
    #pragma unroll
    for (int v = 0; v < 8; ++v)
        localP[(rowBase + v) * N_SPACE + outCol] = acc[v];
}

// ---------------------------------------------------------------------------
// Kernel 2: inter-chunk carry scan.  carry[c,s] = P[c*64, s].
//   E[c,s] = e^{-b} * (local[c*64+63, s] + obs[c*64+63, s])
//   carry[c+1] = carry[c]*e^{-64b} + E[c]
// One block of 256 threads, thread = column s, 64 coalesced steps.
// ---------------------------------------------------------------------------
__global__ __launch_bounds__(256) void hawkes_carry_scan(
    const int* __restrict__ obs,
    const float* __restrict__ localP,
    const float* __restrict__ beta,
    float* __restrict__ carry)        // [NCHUNK, N_SPACE]
{
    const int s = threadIdx.x;
    const float b   = beta[0];
    const float em1 = expf(-b);
    const float emC = expf(-b * (float)CHUNK);

    float c = 0.f;
    for (int ch = 0; ch < NCHUNK; ++ch) {
        carry[ch * N_SPACE + s] = c;
        const int last = ch * CHUNK + CHUNK - 1;
        const float E = em1 * (localP[last * N_SPACE + s] +
                               (float)obs[last * N_SPACE + s]);
        c = c * emC + E;
    }
}

// ---------------------------------------------------------------------------
// Kernel 3: per-query gather + intensity.
//   P(t,s) = local[t,s] + carry[t>>6, s] * e^{-b*(t&63)}
//   lam    = max(mu[s] + alpha[s,s]*b*P, 1e-5)
// ---------------------------------------------------------------------------
__global__ __launch_bounds__(256) void hawkes_gather(
    const int* __restrict__ t,
    const int* __restrict__ s,
    const float* __restrict__ mu,
    const float* __restrict__ alpha,
    const float* __restrict__ beta,
    const float* __restrict__ localP,
    const float* __restrict__ carry,
    float* __restrict__ out)
{
    const int i = blockIdx.x * blockDim.x + threadIdx.x;
    if (i >= BATCH) return;

    const float b  = beta[0];
    const int   ti = t[i];
    const int   si = s[i];
    const int   c  = ti >> 6;
    const int   r  = ti & (CHUNK - 1);

    const float P = localP[ti * N_SPACE + si] +
                    carry[c * N_SPACE + si] * expf(-b * (float)r);
    const float lam = mu[si] + alpha[si * N_SPACE + si] * b * P;
    out[i] = fmaxf(lam, 1e-5f);
}

// ---------------------------------------------------------------------------
// Launch: d_in = {t, s, obs, mu, alpha, beta}; out = float[BATCH].
// Workspace: localP (4 MB) + carry (64 KB).
// ---------------------------------------------------------------------------
extern "C" void kernel_launch(void* const* d_in, const int* in_sizes, int n_in,
                              void* d_out, int out_size, void* d_ws, size_t ws_size,
                              hipStream_t stream) {
    const int*   t     = (const int*)  d_in[0];
    const int*   s     = (const int*)  d_in[1];
    const int*   obs   = (const int*)  d_in[2];
    const float* mu    = (const float*)d_in[3];
    const float* alpha = (const float*)d_in[4];
    const float* beta  = (const float*)d_in[5];
    float*       out   = (float*)d_out;

    float* localP = (float*)d_ws;                               // 4096*256 f32
    float* carry  = localP + (size_t)N_TIME * N_SPACE;          // 64*256 f32

    dim3 g1(NCHUNK, N_SPACE / 64);
    hipLaunchKernelGGL(hawkes_chunk_gemm, g1, dim3(512), 0, stream,
                       obs, beta, localP);

    hipLaunchKernelGGL(hawkes_carry_scan, dim3(1), dim3(N_SPACE), 0, stream,
                       obs, localP, beta, carry);

    hipLaunchKernelGGL(hawkes_gather, dim3(BATCH / 256), dim3(256), 0, stream,
                       t, s, mu, alpha, beta, localP, carry, out);
}